// MAPLoss_87454124081566
// MI455X (gfx1250) — compile-verified
//
#include <hip/hip_runtime.h>

#ifndef __has_builtin
#define __has_builtin(x) 0
#endif

// Problem constants (B=16, C=3, M=N=512)
#define MN_        262144          // 512*512 = 2^18
#define PIX_TOTAL  4194304         // 16 * MN_
#define TILE       256
#define NTILES     (PIX_TOTAL / TILE)   // 16384
#define NBLOCKS    4096
#define EPS_       1e-6f
#define T1_CLIP_   1e7f
#define NEG_INF_   (-3.402823466e38f)

#if __has_builtin(__builtin_amdgcn_global_load_async_to_lds_b128)
#define USE_ASYNC_LDS 1
#else
#define USE_ASYNC_LDS 0
#endif

// Builtin-facing pointer types: param 1 is int4 in global AS (prints as
// "int __vector(4) __device__ *"), param 2 is the LDS-side int4 pointer.
typedef int v4i __attribute__((ext_vector_type(4)));
typedef __attribute__((address_space(1))) v4i* g_v4i_ptr;
typedef __attribute__((address_space(3))) v4i* l_v4i_ptr;

static __device__ __forceinline__ void wait_asynccnt0() {
#if __has_builtin(__builtin_amdgcn_s_wait_asynccnt)
    __builtin_amdgcn_s_wait_asynccnt(0);
#else
    asm volatile("s_wait_asynccnt 0" ::: "memory");
#endif
}

__global__ __launch_bounds__(256) void map_loss_stage1(
    const float* __restrict__ target,
    const float* __restrict__ mu,
    const float* __restrict__ sigma,          // [B,M,N,3,3] -> 9 floats per pixel
    float* __restrict__ wsum,
    float* __restrict__ wmax)
{
    __shared__ __align__(16) float sS[TILE * 9];   // 9216 B sigma tile
    __shared__ float rsum[TILE];
    __shared__ float rmax[TILE];

    const int tid = threadIdx.x;
    float accSum = 0.0f;
    float accMax = NEG_INF_;

    for (int tile = (int)blockIdx.x; tile < NTILES; tile += NBLOCKS) {
        // ---- stage sigma tile (256 pixels * 9 f32 = 576 x b128) into LDS ----
        const float4* gs = (const float4*)(sigma + (size_t)tile * (TILE * 9));
#if USE_ASYNC_LDS
        {
            float4* l4 = (float4*)sS;
            for (int j = tid; j < TILE * 9 / 4; j += TILE) {
                __builtin_amdgcn_global_load_async_to_lds_b128(
                    (g_v4i_ptr)(gs + j),
                    (l_v4i_ptr)(l4 + j),
                    /*offset=*/0, /*cpol=*/0);
            }
            wait_asynccnt0();
        }
#else
        {
            float4* l4 = (float4*)sS;
            for (int j = tid; j < TILE * 9 / 4; j += TILE) l4[j] = gs[j];
        }
#endif
        __syncthreads();

        // ---- per-pixel math ----
        const int p   = tile * TILE + tid;
        const int b   = p >> 18;                 // MN_ = 2^18
        const int pix = p & (MN_ - 1);
        const size_t tb = ((size_t)b * 3) * (size_t)MN_ + (size_t)pix;

        const float t0 = target[tb]             - mu[tb];
        const float t1 = target[tb + MN_]       - mu[tb + MN_];
        const float t2 = target[tb + 2 * MN_]   - mu[tb + 2 * MN_];

        const float* S = &sS[tid * 9];           // stride 9 across lanes: conflict-free
        const float s00 = S[0], s01 = S[1], s02 = S[2];
        const float s10 = S[3], s11 = S[4], s12 = S[5];
        const float s20 = S[6], s21 = S[7], s22 = S[8];

        // cofactors c[i][j] of S; inv[i][j] = c[j][i] / det
        const float c00 = s11 * s22 - s12 * s21;
        const float c01 = s12 * s20 - s10 * s22;
        const float c02 = s10 * s21 - s11 * s20;
        const float c10 = s02 * s21 - s01 * s22;
        const float c11 = s00 * s22 - s02 * s20;
        const float c12 = s01 * s20 - s00 * s21;
        const float c20 = s01 * s12 - s02 * s11;
        const float c21 = s02 * s10 - s00 * s12;
        const float c22 = s00 * s11 - s01 * s10;

        const float det = s00 * c00 + s01 * c01 + s02 * c02;

        const float q =
            t0 * (c00 * t0 + c10 * t1 + c20 * t2) +
            t1 * (c01 * t0 + c11 * t1 + c21 * t2) +
            t2 * (c02 * t0 + c12 * t1 + c22 * t2);

        const float t1v  = 0.5f * q / det;
        const float dets = 0.5f * logf(fmaxf(det, EPS_));

        accSum += t1v + dets;
        accMax  = fmaxf(accMax, t1v);
        __syncthreads();   // protect LDS before next tile's async copy
    }

    // ---- block reduction ----
    rsum[tid] = accSum;
    rmax[tid] = accMax;
    __syncthreads();
    for (int s = TILE / 2; s > 0; s >>= 1) {
        if (tid < s) {
            rsum[tid] += rsum[tid + s];
            rmax[tid] = fmaxf(rmax[tid], rmax[tid + s]);
        }
        __syncthreads();
    }
    if (tid == 0) {
        wsum[blockIdx.x] = rsum[0];
        wmax[blockIdx.x] = rmax[0];
    }
}

__global__ __launch_bounds__(256) void map_loss_stage2(
    const float* __restrict__ wsum,
    const float* __restrict__ wmax,
    float* __restrict__ out)
{
    __shared__ double dsum[256];
    __shared__ float  dmax[256];
    const int tid = threadIdx.x;

    double s = 0.0;
    float  m = NEG_INF_;
    for (int i = tid; i < NBLOCKS; i += 256) {
        s += (double)wsum[i];
        m  = fmaxf(m, wmax[i]);
    }
    dsum[tid] = s;
    dmax[tid] = m;
    __syncthreads();
    for (int k = 128; k > 0; k >>= 1) {
        if (tid < k) {
            dsum[tid] += dsum[tid + k];
            dmax[tid] = fmaxf(dmax[tid], dmax[tid + k]);
        }
        __syncthreads();
    }
    if (tid == 0) {
        const float loss = (float)(dsum[0] * (1.0 / (double)PIX_TOTAL));
        out[0] = (dmax[0] > T1_CLIP_) ? 0.0f : loss;
    }
}

extern "C" void kernel_launch(void* const* d_in, const int* in_sizes, int n_in,
                              void* d_out, int out_size, void* d_ws, size_t ws_size,
                              hipStream_t stream) {
    (void)in_sizes; (void)n_in; (void)out_size; (void)ws_size;
    // setup_inputs order: target, mu, sigma_mu(unused), sigma_n(unused), sigma_y
    const float* target  = (const float*)d_in[0];
    const float* mu      = (const float*)d_in[1];
    const float* sigma_y = (const float*)d_in[4];
    float* out  = (float*)d_out;
    float* wsum = (float*)d_ws;            // NBLOCKS floats
    float* wmax = wsum + NBLOCKS;          // NBLOCKS floats (32 KB total)

    map_loss_stage1<<<NBLOCKS, TILE, 0, stream>>>(target, mu, sigma_y, wsum, wmax);
    map_loss_stage2<<<1, 256, 0, stream>>>(wsum, wmax, out);
}